// HeteroGNN_4681514352901
// MI455X (gfx1250) — compile-verified
//
#include <hip/hip_runtime.h>

#define HN 50000
#define FN 200000
#define DD 128
#define DO 32
#define EE 1000000
#define LDS_STRIDE 132   // 128 + 4 pad -> conflict-free column reads (132 % 64 == 4)

typedef __attribute__((ext_vector_type(2))) float v2f;
typedef __attribute__((ext_vector_type(8))) float v8f;

// ---------------------------------------------------------------- utilities
__global__ __launch_bounds__(256) void k_zero(float* __restrict__ p, int n4) {
  int i = blockIdx.x * blockDim.x + threadIdx.x;
  int stride = gridDim.x * blockDim.x;
  float4 z = make_float4(0.f, 0.f, 0.f, 0.f);
  for (; i < n4; i += stride) ((float4*)p)[i] = z;
}

__global__ __launch_bounds__(256) void k_degree(const int* __restrict__ dst,
                                                float* cnt) {
  int i = blockIdx.x * blockDim.x + threadIdx.x;
  if (i < EE) atomicAdd(&cnt[dst[i]], 1.0f);
}

__global__ __launch_bounds__(256) void k_invert(float* c, int n) {
  int i = blockIdx.x * blockDim.x + threadIdx.x;
  if (i < n) c[i] = 1.0f / fmaxf(c[i], 1.0f);
}

// one wave32 per edge: gather a 128-float row, atomic-add into agg[dst]
__global__ __launch_bounds__(256) void k_scatter(const float* __restrict__ xsrc,
                                                 const int* __restrict__ src,
                                                 const int* __restrict__ dst,
                                                 float* agg) {
  int e    = (blockIdx.x * blockDim.x + threadIdx.x) >> 5;
  int lane = threadIdx.x & 31;
  if (e >= EE) return;
  int s = src[e];
  int d = dst[e];
  float4 v = *(const float4*)(xsrc + (size_t)s * DD + lane * 4);
  float* a = agg + (size_t)d * DD + lane * 4;
  atomicAdd(a + 0, v.x);
  atomicAdd(a + 1, v.y);
  atomicAdd(a + 2, v.z);
  atomicAdd(a + 3, v.w);
}

// ------------------------------------------------- fused SAGE layer (D=128)
// out[m,n] = leakyrelu( (A1[m,:]*scale[m]) @ W1[:,n] + X[m,:] @ W2[:,n]
//                       + b1[n] + b2[n] )
// Block = 8 waves = 128 output cols; block computes a 32-row slab. The two
// activation operands are staged (mean-scale pre-folded) into padded LDS once
// per block, then each wave runs two 16x16 accumulator chains against one
// shared B fragment per k-step (V_WMMA_F32_16X16X4_F32). `out` may alias `X`:
// all X reads happen in the staging pass before the barrier, stores after.
__global__ __launch_bounds__(256) void k_sage(const float* __restrict__ A1,
                                              const float* __restrict__ scale,
                                              const float* X,
                                              const float* __restrict__ W1,
                                              const float* __restrict__ W2,
                                              const float* __restrict__ b1,
                                              const float* __restrict__ b2,
                                              float* out, int nrows) {
  __shared__ float sA[32 * LDS_STRIDE];
  __shared__ float sX[32 * LDS_STRIDE];

  const int lane = threadIdx.x & 31;
  const int l16  = lane & 15;
  const int kh   = lane >> 4;                 // 0: K pair {0,1}, 1: {2,3}
  const int n0   = (threadIdx.x >> 5) * 16;   // wave -> 16-col tile
  const int m0   = blockIdx.x * 32;
  const bool hasHi = (m0 + 16) < nrows;       // block-uniform

  // ---- stage 32 activation rows into LDS (scale folded into A1) ----
  for (int c = threadIdx.x; c < (32 * DD) / 4; c += 256) {
    const int row = c >> 5;                   // 32 float4-chunks per row
    const int col = (c & 31) * 4;
    int gr = m0 + row;
    gr = (gr < nrows) ? gr : (nrows - 1);     // clamp tail (tile-1 not stored)
    const float s  = scale[gr];
    const float4 va = *(const float4*)(A1 + (size_t)gr * DD + col);
    const float4 vx = *(const float4*)(X  + (size_t)gr * DD + col);
    float* pa = &sA[row * LDS_STRIDE + col];
    float* px = &sX[row * LDS_STRIDE + col];
    pa[0] = va.x * s; pa[1] = va.y * s; pa[2] = va.z * s; pa[3] = va.w * s;
    px[0] = vx.x;     px[1] = vx.y;     px[2] = vx.z;     px[3] = vx.w;
  }
  __syncthreads();

  v8f acc0 = {};
  v8f acc1 = {};

#pragma unroll
  for (int k = 0; k < DD; k += 4) {
    const int ka = k + 2 * kh;
    v2f b;
    b.x = W1[(size_t)ka * DD + n0 + l16];
    b.y = W1[(size_t)(ka + 1) * DD + n0 + l16];
    v2f a0 = *(const v2f*)(&sA[l16 * LDS_STRIDE + ka]);
    v2f a1 = *(const v2f*)(&sA[(16 + l16) * LDS_STRIDE + ka]);
    acc0 = __builtin_amdgcn_wmma_f32_16x16x4_f32(false, a0, false, b,
                                                 (short)0, acc0, false, false);
    acc1 = __builtin_amdgcn_wmma_f32_16x16x4_f32(false, a1, false, b,
                                                 (short)0, acc1, false, false);
  }
#pragma unroll
  for (int k = 0; k < DD; k += 4) {
    const int ka = k + 2 * kh;
    v2f b;
    b.x = W2[(size_t)ka * DD + n0 + l16];
    b.y = W2[(size_t)(ka + 1) * DD + n0 + l16];
    v2f a0 = *(const v2f*)(&sX[l16 * LDS_STRIDE + ka]);
    v2f a1 = *(const v2f*)(&sX[(16 + l16) * LDS_STRIDE + ka]);
    acc0 = __builtin_amdgcn_wmma_f32_16x16x4_f32(false, a0, false, b,
                                                 (short)0, acc0, false, false);
    acc1 = __builtin_amdgcn_wmma_f32_16x16x4_f32(false, a1, false, b,
                                                 (short)0, acc1, false, false);
  }

  const float bias  = b1[n0 + l16] + b2[n0 + l16];
  const int   rbase = m0 + 8 * kh;            // C/D: vgpr v -> row v (+8 hi)
#pragma unroll
  for (int v = 0; v < 8; ++v) {
    float val = acc0[v] + bias;
    val = (val > 0.f) ? val : 0.01f * val;    // leaky_relu(0.01)
    out[(size_t)(rbase + v) * DD + n0 + l16] = val;
  }
  if (hasHi) {
#pragma unroll
    for (int v = 0; v < 8; ++v) {
      float val = acc1[v] + bias;
      val = (val > 0.f) ? val : 0.01f * val;
      out[(size_t)(rbase + 16 + v) * DD + n0 + l16] = val;
    }
  }
}

// ------------------------------------------- output projection [N,128]x[128,32]
__global__ __launch_bounds__(64) void k_out(const float* __restrict__ X,
                                            const float* __restrict__ W,
                                            const float* __restrict__ b,
                                            float* __restrict__ out) {
  const int lane = threadIdx.x & 31;
  const int l16  = lane & 15;
  const int kh   = lane >> 4;
  const int n0   = (threadIdx.x >> 5) * 16;   // 2 waves -> 32 cols
  const int m0   = blockIdx.x * 16;
  const size_t rowA = (size_t)(m0 + l16) * DD;

  v8f acc = {};
#pragma unroll
  for (int k = 0; k < DD; k += 4) {
    const int ka = k + 2 * kh;
    v2f a, bb;
    a.x  = X[rowA + ka];
    a.y  = X[rowA + ka + 1];
    bb.x = W[(size_t)ka * DO + n0 + l16];
    bb.y = W[(size_t)(ka + 1) * DO + n0 + l16];
    acc = __builtin_amdgcn_wmma_f32_16x16x4_f32(false, a, false, bb,
                                                (short)0, acc, false, false);
  }
  const float bias  = b[n0 + l16];
  const int   rbase = m0 + 8 * kh;
#pragma unroll
  for (int v = 0; v < 8; ++v)
    out[(size_t)(rbase + v) * DO + n0 + l16] = acc[v] + bias;
}

// ---------------------------------------------------------------- dispatcher
extern "C" void kernel_launch(void* const* d_in, const int* in_sizes, int n_in,
                              void* d_out, int out_size, void* d_ws, size_t ws_size,
                              hipStream_t stream) {
  const float* x_host  = (const float*)d_in[0];
  const float* x_flow  = (const float*)d_in[1];
  const int*   ehf_src = (const int*)d_in[2];
  const int*   ehf_dst = (const int*)d_in[3];
  const int*   efh_src = (const int*)d_in[4];
  const int*   efh_dst = (const int*)d_in[5];
  const float* Wl      = (const float*)d_in[6];   // (L,2,D,D)
  const float* bl      = (const float*)d_in[7];   // (L,2,D)
  const float* Wr      = (const float*)d_in[8];
  const float* br      = (const float*)d_in[9];
  const float* W_out   = (const float*)d_in[10];  // (D, D_OUT)
  const float* b_out   = (const float*)d_in[11];
  float*       out     = (float*)d_out;

  float* ws  = (float*)d_ws;
  size_t off = 0;
  float* f1    = ws + off; off += (size_t)FN * DD;   // flow features
  float* h1    = ws + off; off += (size_t)HN * DD;   // host features
  float* agg_f = ws + off; off += (size_t)FN * DD;   // flow aggregation (102MB, L2-resident)
  float* agg_h = ws + off; off += (size_t)HN * DD;   // host aggregation
  float* icf   = ws + off; off += FN;                // 1/max(deg,1) flow
  float* ich   = ws + off; off += HN;                // 1/max(deg,1) host

  const int ZB = 2048;                      // grid-stride zero blocks
  const int EB = (EE + 255) / 256;          // 1 thread / edge
  const int SB = (EE * 32 + 255) / 256;     // 1 wave / edge
  const int GF = (FN + 31) / 32;            // flow sage blocks (exact)
  const int GH = (HN + 31) / 32;            // host sage blocks (1 tail tile)

  // in-degree -> reciprocal (edge lists are layer-invariant)
  k_zero<<<ZB, 256, 0, stream>>>(icf, FN / 4);
  k_zero<<<ZB, 256, 0, stream>>>(ich, HN / 4);
  k_degree<<<EB, 256, 0, stream>>>(ehf_dst, icf);
  k_degree<<<EB, 256, 0, stream>>>(efh_dst, ich);
  k_invert<<<(FN + 255) / 256, 256, 0, stream>>>(icf, FN);
  k_invert<<<(HN + 255) / 256, 256, 0, stream>>>(ich, HN);

  // ---- layer 0 ----
  k_zero<<<ZB, 256, 0, stream>>>(agg_f, FN * DD / 4);
  k_zero<<<ZB, 256, 0, stream>>>(agg_h, HN * DD / 4);
  k_scatter<<<SB, 256, 0, stream>>>(x_host, ehf_src, ehf_dst, agg_f);
  k_scatter<<<SB, 256, 0, stream>>>(x_flow, efh_src, efh_dst, agg_h);
  k_sage<<<GF, 256, 0, stream>>>(agg_f, icf, x_flow,
                                 Wl + 0 * DD * DD, Wr + 0 * DD * DD,
                                 bl + 0 * DD, br + 0 * DD, f1, FN);
  k_sage<<<GH, 256, 0, stream>>>(agg_h, ich, x_host,
                                 Wl + 1 * DD * DD, Wr + 1 * DD * DD,
                                 bl + 1 * DD, br + 1 * DD, h1, HN);

  // ---- layer 1 (host-side update is dead: output only consumes f) ----
  k_zero<<<ZB, 256, 0, stream>>>(agg_f, FN * DD / 4);
  k_scatter<<<SB, 256, 0, stream>>>(h1, ehf_src, ehf_dst, agg_f);
  k_sage<<<GF, 256, 0, stream>>>(agg_f, icf, f1,
                                 Wl + 2 * DD * DD, Wr + 2 * DD * DD,
                                 bl + 2 * DD, br + 2 * DD, f1, FN);  // in-place

  // ---- output projection ----
  k_out<<<FN / 16, 64, 0, stream>>>(f1, W_out, b_out, out);
}